// MergedEmbeddingBag_6708738916560
// MI455X (gfx1250) — compile-verified
//
#include <hip/hip_runtime.h>

// Problem constants (fixed by the reference harness)
#define T_TABLES   8
#define N_ROWS     100000
#define EMB_D      128
#define N_BAGS     16384
#define TOTAL_IDX  (N_BAGS * 20)

#define WAVES_PER_BLOCK 8
#define THREADS         (WAVES_PER_BLOCK * 32)
#define DEPTH           4            // async rows in flight per wave
#define NBUF            8            // 2*DEPTH buffers -> no WAR on buffer reuse

typedef __attribute__((ext_vector_type(4))) float v4f;

__global__ __launch_bounds__(THREADS)
void merged_embbag_fwd(const float* __restrict__ weights,
                       const int*   __restrict__ indices,
                       const int*   __restrict__ offsets,
                       float*       __restrict__ out)
{
    // Per-wave row staging: NBUF buffers of one 512B embedding row each.
    __shared__ v4f smem[WAVES_PER_BLOCK][NBUF][32];

    const int lane = threadIdx.x & 31;
    // Force wave-uniformity so everything derived from it goes down the
    // SALU/SMEM (constant cache) path instead of divergent VMEM.
    const int wave = __builtin_amdgcn_readfirstlane(threadIdx.x >> 5);

    const int bag_global = __builtin_amdgcn_readfirstlane(blockIdx.x) * WAVES_PER_BLOCK + wave;
    const int t = bag_global / N_BAGS;                             // t-major grid
    const int b = bag_global - t * N_BAGS;

    const int*   idx  = indices + (size_t)t * TOTAL_IDX;
    const int*   off  = offsets + (size_t)t * N_BAGS;
    const float* wtab = weights + (size_t)t * N_ROWS * EMB_D;

    const int start = __builtin_amdgcn_readfirstlane(off[b]);      // s_load
    const int end   = (b + 1 < N_BAGS)
                        ? __builtin_amdgcn_readfirstlane(off[b + 1])
                        : TOTAL_IDX;
    const int count = end - start;

    // Issue async gather of row j into LDS buffer (j % NBUF).
    auto issue = [&](int j) {
        const int row = __builtin_amdgcn_readfirstlane(idx[start + j]);  // s_load
        const float* src = wtab + (size_t)row * EMB_D + lane * 4;  // 512B/row coalesced
        unsigned lds = (unsigned)(unsigned long long)&smem[wave][j & (NBUF - 1)][lane];
        asm volatile("global_load_async_to_lds_b128 %0, %1, off"
                     :: "v"(lds), "v"(src)
                     : "memory");
    };

    v4f acc = (v4f)(0.f);

    const int pre = count < DEPTH ? count : DEPTH;
    for (int j = 0; j < pre; ++j) issue(j);

    int j = 0;
    for (; j + DEPTH < count; ++j) {
        issue(j + DEPTH);                       // refill pipeline (different buffer than row j)
        // In-order async completion: after issuing j+DEPTH+1 loads,
        // ASYNCcnt<=DEPTH guarantees row j has landed in LDS.
        asm volatile("s_wait_asynccnt 4" ::: "memory");
        v4f v = smem[wave][j & (NBUF - 1)][lane];
        acc += v;
    }
    asm volatile("s_wait_asynccnt 0" ::: "memory");
    for (; j < count; ++j) {
        v4f v = smem[wave][j & (NBUF - 1)][lane];
        acc += v;
    }

    // Streaming, write-once output: non-temporal so we don't evict hot
    // embedding rows from the 192MB L2.
    v4f* dst = (v4f*)(out + ((size_t)t * N_BAGS + b) * EMB_D) + lane;
    __builtin_nontemporal_store(acc, dst);
}

extern "C" void kernel_launch(void* const* d_in, const int* in_sizes, int n_in,
                              void* d_out, int out_size, void* d_ws, size_t ws_size,
                              hipStream_t stream)
{
    (void)in_sizes; (void)n_in; (void)out_size; (void)d_ws; (void)ws_size;
    const float* weights = (const float*)d_in[0];   // [T, N, D] f32
    const int*   indices = (const int*)d_in[1];     // [T, TOTAL] i32
    const int*   offsets = (const int*)d_in[2];     // [T, B] i32
    float*       out     = (float*)d_out;           // [T, B, D] f32

    const int total_bags = T_TABLES * N_BAGS;       // 131072 waves
    const int blocks     = total_bags / WAVES_PER_BLOCK;
    hipLaunchKernelGGL(merged_embbag_fwd, dim3(blocks), dim3(THREADS), 0, stream,
                       weights, indices, offsets, out);
}